// GATNet_51204600103281
// MI455X (gfx1250) — compile-verified
//
#include <hip/hip_runtime.h>
#include <hip/hip_bf16.h>
#include <float.h>

#define N_NODES 50000
#define N_EDGES 800000
#define N_GRAPH 512
#define IN_DIM  4
#define HEADS   4
#define CH      64
#define HC      256       // HEADS*CH
#define NCLS    2
#define BN_EPS  1e-5f
#define NEG_SLOPE 0.2f

typedef float v2f __attribute__((ext_vector_type(2)));
typedef float v8f __attribute__((ext_vector_type(8)));

// ---------------------------------------------------------------- utilities
__device__ __forceinline__ void atomicMaxFloat(float* addr, float v) {
    // int ordering == float ordering for non-negative floats;
    // uint ordering is reversed for negative floats. Init must be -inf.
    if (v >= 0.0f) atomicMax((int*)addr, __float_as_int(v));
    else           atomicMin((unsigned int*)addr, __float_as_uint(v));
}

__global__ void k_fill(float* __restrict__ p, float v, long n) {
    long i = (long)blockIdx.x * blockDim.x + threadIdx.x;
    if (i < n) p[i] = v;
}

// ---------------------------------------------------------------- input BN
__global__ void k_bn_in(const float* __restrict__ x,
                        const float* __restrict__ gamma, const float* __restrict__ beta,
                        const float* __restrict__ mean,  const float* __restrict__ var,
                        float* __restrict__ y, int total, int c) {
    int i = blockIdx.x * blockDim.x + threadIdx.x;
    if (i >= total) return;
    int j = i % c;
    y[i] = (x[i] - mean[j]) * (gamma[j] * rsqrtf(var[j] + BN_EPS)) + beta[j];
}

// ---------------------------------------------------------------- WMMA GEMM
// D[M,Nc] = A[M,K] @ B[K,Nc]  (+bias, ReLU if relu_bias). One wave per 16x16
// tile using V_WMMA_F32_16X16X4_F32. Requires M%16==0, Nc%16==0, K%4==0.
__global__ void k_gemm_wmma(const float* __restrict__ A, const float* __restrict__ B,
                            const float* __restrict__ bias, float* __restrict__ D,
                            int M, int K, int Nc, int relu_bias) {
    int wave = (blockIdx.x * blockDim.x + threadIdx.x) >> 5;
    int lane = threadIdx.x & 31;
    int ntiles = Nc >> 4;
    int mt = wave / ntiles;
    int nt = wave - mt * ntiles;
    if (mt * 16 >= M) return;                 // wave-uniform: EXEC stays all-1s
    int half = lane >> 4;                     // 0 or 1
    int l16  = lane & 15;

    // A: lane l16 = row, VGPR pair holds K = 2*half + {0,1}
    const float* arow = A + (size_t)(mt * 16 + l16) * K + 2 * half;
    // B: lane l16 = col, VGPR pair holds rows K = 2*half + {0,1}
    const float* bcol = B + (size_t)(2 * half) * Nc + (nt * 16 + l16);

    v8f acc = {0.f, 0.f, 0.f, 0.f, 0.f, 0.f, 0.f, 0.f};
    for (int k0 = 0; k0 < K; k0 += 4) {
        v2f a = *(const v2f*)(arow + k0);     // 8B-aligned: K even, k0%4==0
        v2f b;
        b.x = bcol[(size_t)k0 * Nc];
        b.y = bcol[(size_t)(k0 + 1) * Nc];
        acc = __builtin_amdgcn_wmma_f32_16x16x4_f32(
            false, a, false, b, (short)0, acc, false, false);
    }

    int col = nt * 16 + l16;
    float bv = relu_bias ? bias[col] : 0.f;
    #pragma unroll
    for (int r = 0; r < 8; ++r) {
        int row = mt * 16 + r + 8 * half;
        float v = acc[r] + bv;
        if (relu_bias) v = fmaxf(v, 0.f);
        D[(size_t)row * Nc + col] = v;
    }
}

// ---------------------------------------------------------------- attention
// a_src[n,h] = sum_c h[n,h,c]*att_src[h,c]; same for a_dst
__global__ void k_attn(const float* __restrict__ h,
                       const float* __restrict__ att_src, const float* __restrict__ att_dst,
                       float* __restrict__ a_src, float* __restrict__ a_dst, int n) {
    int i = blockIdx.x * blockDim.x + threadIdx.x;
    if (i >= n * HEADS) return;
    int hh = i & (HEADS - 1);
    int node = i >> 2;
    const float* hp = h + (size_t)node * HC + hh * CH;
    const float* as = att_src + hh * CH;
    const float* ad = att_dst + hh * CH;
    float s = 0.f, d = 0.f;
    #pragma unroll 8
    for (int c = 0; c < CH; ++c) { float v = hp[c]; s += v * as[c]; d += v * ad[c]; }
    a_src[i] = s;
    a_dst[i] = d;
}

// ---------------------------------------------------------------- edge pass 1: segment max
__global__ void k_edge_max(const int* __restrict__ src, const int* __restrict__ dst,
                           const float* __restrict__ a_src, const float* __restrict__ a_dst,
                           float* __restrict__ seg_max, int e) {
    int i = blockIdx.x * blockDim.x + threadIdx.x;
    if (i >= e * HEADS) return;
    int hh = i & (HEADS - 1);
    int ed = i >> 2;
    int s = src[ed], d = dst[ed];
    float v = a_src[s * HEADS + hh] + a_dst[d * HEADS + hh];
    v = v > 0.f ? v : NEG_SLOPE * v;                // leaky relu
    atomicMaxFloat(&seg_max[d * HEADS + hh], v);
}

// ---------------------------------------------------------------- edge pass 2: exp + segment sum
__global__ void k_edge_expsum(const int* __restrict__ src, const int* __restrict__ dst,
                              const float* __restrict__ a_src, const float* __restrict__ a_dst,
                              const float* __restrict__ seg_max,
                              float* __restrict__ alpha, float* __restrict__ seg_sum, int e) {
    int i = blockIdx.x * blockDim.x + threadIdx.x;
    if (i >= e * HEADS) return;
    int hh = i & (HEADS - 1);
    int ed = i >> 2;
    int s = src[ed], d = dst[ed];
    float v = a_src[s * HEADS + hh] + a_dst[d * HEADS + hh];
    v = v > 0.f ? v : NEG_SLOPE * v;
    float ex = __expf(v - seg_max[d * HEADS + hh]);
    alpha[i] = ex;
    atomicAdd(&seg_sum[d * HEADS + hh], ex);
}

// ---------------------------------------------------------------- edge pass 3: weighted scatter-add
// one block (256 threads = full HC) per edge; contiguous loads/atomics (L2-resident target)
__global__ void k_edge_agg(const int* __restrict__ src, const int* __restrict__ dst,
                           const float* __restrict__ h, const float* __restrict__ alpha,
                           const float* __restrict__ seg_sum, float* __restrict__ outagg) {
    int e = blockIdx.x;
    int t = threadIdx.x;          // 0..255 -> (head = t>>6, channel = t&63)
    int hh = t >> 6;
    int s = src[e], d = dst[e];
    float al = alpha[e * HEADS + hh] / (seg_sum[d * HEADS + hh] + 1e-16f);
    atomicAdd(&outagg[(size_t)d * HC + t], h[(size_t)s * HC + t] * al);
}

// ---------------------------------------------------------------- bias + BN + ReLU
__global__ void k_bias_bn_relu(const float* __restrict__ in, const float* __restrict__ bias,
                               const float* __restrict__ gamma, const float* __restrict__ beta,
                               const float* __restrict__ mean, const float* __restrict__ var,
                               float* __restrict__ out, int total) {
    int i = blockIdx.x * blockDim.x + threadIdx.x;
    if (i >= total) return;
    int j = i & (HC - 1);
    float v = in[i] + bias[j];
    v = (v - mean[j]) * (gamma[j] * rsqrtf(var[j] + BN_EPS)) + beta[j];
    out[i] = fmaxf(v, 0.f);
}

// ---------------------------------------------------------------- mean pool
__global__ void k_pool(const int* __restrict__ batch, const float* __restrict__ xf,
                       float* __restrict__ pool_sum, float* __restrict__ pool_cnt) {
    int n = blockIdx.x;
    int c = threadIdx.x;          // 0..63
    int g = batch[n];
    atomicAdd(&pool_sum[(size_t)g * CH + c], xf[(size_t)n * CH + c]);
    if (c == 0) atomicAdd(&pool_cnt[g], 1.0f);
}

__global__ void k_pool_div(const float* __restrict__ pool_sum, const float* __restrict__ pool_cnt,
                           float* __restrict__ g, int total) {
    int i = blockIdx.x * blockDim.x + threadIdx.x;
    if (i >= total) return;
    g[i] = pool_sum[i] / fmaxf(pool_cnt[i / CH], 1.0f);
}

// ---------------------------------------------------------------- final tiny classifier GEMM (Nc=2)
__global__ void k_cls2(const float* __restrict__ h1, const float* __restrict__ W2,
                       const float* __restrict__ b2, float* __restrict__ out, int total) {
    int i = blockIdx.x * blockDim.x + threadIdx.x;
    if (i >= total) return;
    int g = i / NCLS, c = i % NCLS;
    float acc = b2[c];
    #pragma unroll 8
    for (int k = 0; k < CH; ++k) acc += h1[(size_t)g * CH + k] * W2[k * NCLS + c];
    out[i] = acc;
}

// =================================================================
extern "C" void kernel_launch(void* const* d_in, const int* in_sizes, int n_in,
                              void* d_out, int out_size, void* d_ws, size_t ws_size,
                              hipStream_t stream) {
    (void)in_sizes; (void)n_in; (void)out_size; (void)ws_size;
    const int N = N_NODES, E = N_EDGES, G = N_GRAPH;

    // --- inputs: setup_inputs() dict order, params flattened by sorted keys ---
    const float* x     = (const float*)d_in[0];          // [N,4]
    const int*   eidx  = (const int*)d_in[1];            // [2,E] (src row 0, dst row 1)
    const int*   batch = (const int*)d_in[2];            // [N]
    const float* cls_W1  = (const float*)d_in[3];        // [64,64]
    const float* cls_W2  = (const float*)d_in[4];        // [64,2]
    const float* cls_b1  = (const float*)d_in[5];        // [64]
    const float* cls_b2  = (const float*)d_in[6];        // [2]
    const float* final_W = (const float*)d_in[7];        // [256,64]
    const float* final_b = (const float*)d_in[8];        // [64]
    const float* bn_beta_in  = (const float*)d_in[9];    // in_bn: beta,gamma,mean,var (sorted)
    const float* bn_gamma_in = (const float*)d_in[10];
    const float* bn_mean_in  = (const float*)d_in[11];
    const float* bn_var_in   = (const float*)d_in[12];
    // layers[i] leaves (sorted): W, att_dst, att_src, bias, bn_beta, bn_gamma, bn_mean, bn_var
    auto LP = [&](int layer, int j) { return (const float*)d_in[13 + 8 * layer + j]; };

    const int* src = eidx;
    const int* dst = eidx + E;

    // --- workspace carve-up (floats) ---
    float* ws = (float*)d_ws;
    size_t o = 0;
    float* x0      = ws + o; o += (size_t)N * IN_DIM;   // BN'd input
    float* hbuf    = ws + o; o += (size_t)N * HC;       // h = x@W
    float* xcur    = ws + o; o += (size_t)N * HC;       // layer output
    float* outagg  = ws + o; o += (size_t)N * HC;       // scatter accumulator
    float* a_src   = ws + o; o += (size_t)N * HEADS;
    float* a_dst   = ws + o; o += (size_t)N * HEADS;
    float* seg_max = ws + o; o += (size_t)N * HEADS;
    float* seg_sum = ws + o; o += (size_t)N * HEADS;
    float* alphaB  = ws + o; o += (size_t)E * HEADS;
    float* pool_s  = ws + o; o += (size_t)G * CH;
    float* pool_c  = ws + o; o += (size_t)G;
    float* gfeat   = ws + o; o += (size_t)G * CH;
    float* h1      = ws + o; o += (size_t)G * CH;
    float* xf      = ws + o; o += (size_t)N * CH;       // final node features

    auto cdiv = [](long a, long b) { return (int)((a + b - 1) / b); };

    // --- input BN ---
    k_bn_in<<<cdiv((long)N * IN_DIM, 256), 256, 0, stream>>>(
        x, bn_gamma_in, bn_beta_in, bn_mean_in, bn_var_in, x0, N * IN_DIM, IN_DIM);

    // --- 3 GAT layers ---
    for (int L = 0; L < 3; ++L) {
        const float* in_ptr = (L == 0) ? x0 : xcur;
        int K = (L == 0) ? IN_DIM : HC;

        // h = in @ W   [N,HC]  (WMMA f32 16x16x4)
        {
            long waves = ((long)N / 16) * (HC / 16);
            k_gemm_wmma<<<cdiv(waves * 32, 256), 256, 0, stream>>>(
                in_ptr, LP(L, 0), nullptr, hbuf, N, K, HC, 0);
        }
        // per-node attention logits
        k_attn<<<cdiv((long)N * HEADS, 256), 256, 0, stream>>>(
            hbuf, LP(L, 2), LP(L, 1), a_src, a_dst, N);

        // init segment buffers + accumulator
        k_fill<<<cdiv((long)N * HEADS, 256), 256, 0, stream>>>(seg_max, -INFINITY, (long)N * HEADS);
        k_fill<<<cdiv((long)N * HEADS, 256), 256, 0, stream>>>(seg_sum, 0.f, (long)N * HEADS);
        k_fill<<<cdiv((long)N * HC, 256), 256, 0, stream>>>(outagg, 0.f, (long)N * HC);

        // segment softmax over edges + weighted scatter
        k_edge_max<<<cdiv((long)E * HEADS, 256), 256, 0, stream>>>(
            src, dst, a_src, a_dst, seg_max, E);
        k_edge_expsum<<<cdiv((long)E * HEADS, 256), 256, 0, stream>>>(
            src, dst, a_src, a_dst, seg_max, alphaB, seg_sum, E);
        k_edge_agg<<<E, HC, 0, stream>>>(src, dst, hbuf, alphaB, seg_sum, outagg);

        // + bias, BN(eval), ReLU
        k_bias_bn_relu<<<cdiv((long)N * HC, 256), 256, 0, stream>>>(
            outagg, LP(L, 3), LP(L, 5), LP(L, 4), LP(L, 6), LP(L, 7), xcur, N * HC);
    }

    // --- final projection: xf = relu(xcur @ final_W + final_b)  [N,64] ---
    {
        long waves = ((long)N / 16) * (CH / 16);
        k_gemm_wmma<<<cdiv(waves * 32, 256), 256, 0, stream>>>(
            xcur, final_W, final_b, xf, N, HC, CH, 1);
    }

    // --- global mean pool per graph ---
    k_fill<<<cdiv((long)G * CH, 256), 256, 0, stream>>>(pool_s, 0.f, (long)G * CH);
    k_fill<<<cdiv((long)G, 256), 256, 0, stream>>>(pool_c, 0.f, (long)G);
    k_pool<<<N, CH, 0, stream>>>(batch, xf, pool_s, pool_c);
    k_pool_div<<<cdiv((long)G * CH, 256), 256, 0, stream>>>(pool_s, pool_c, gfeat, G * CH);

    // --- classifier: h1 = relu(g @ W1 + b1), out = h1 @ W2 + b2 ---
    {
        long waves = ((long)G / 16) * (CH / 16);
        k_gemm_wmma<<<cdiv(waves * 32, 256), 256, 0, stream>>>(
            gfeat, cls_W1, cls_b1, h1, G, CH, CH, 1);
    }
    k_cls2<<<cdiv((long)G * NCLS, 256), 256, 0, stream>>>(h1, cls_W2, cls_b2, (float*)d_out, G * NCLS);
}